// GraphConvolution_3796751090176
// MI455X (gfx1250) — compile-verified
//
#include <hip/hip_runtime.h>

typedef __attribute__((ext_vector_type(2))) float v2f;
typedef __attribute__((ext_vector_type(4))) float v4f;
typedef __attribute__((ext_vector_type(8))) float v8f;
typedef int i32x4 __attribute__((vector_size(16)));

#define U_DIM 4000
#define FIN   96
#define HID   32
#define NCLS  5
#define BK    80      // K tile (4000 % 80 == 0 -> no K tail)
#define KSTR  84      // padded LDS row stride (dwords) -> conflict-free b64 reads
#define WAVES 4
#define ROWS_PER_WG (WAVES * 16)

// ---- CDNA5 async global->LDS copy (ASYNCcnt-tracked), with sync fallback ----
#if defined(__gfx1250__) &&                                                    \
    __has_builtin(__builtin_amdgcn_global_load_async_to_lds_b32) &&            \
    __has_builtin(__builtin_amdgcn_global_load_async_to_lds_b128) &&           \
    __has_builtin(__builtin_amdgcn_s_wait_asynccnt)
// generic LDS address = {aperture, offset[31:0]}; low 32 bits are the AS(3) offset
#define ASYNC_B32(src, dst)                                                    \
    __builtin_amdgcn_global_load_async_to_lds_b32(                             \
        (__attribute__((address_space(1))) int*)(uintptr_t)(src),              \
        (__attribute__((address_space(3))) int*)(unsigned)(uintptr_t)(dst),    \
        0, 0)
#define ASYNC_B128(src, dst)                                                   \
    __builtin_amdgcn_global_load_async_to_lds_b128(                            \
        (__attribute__((address_space(1))) i32x4*)(uintptr_t)(src),            \
        (__attribute__((address_space(3))) i32x4*)(unsigned)(uintptr_t)(dst),  \
        0, 0)
#define ASYNC_WAIT() __builtin_amdgcn_s_wait_asynccnt(0)
#else
#define ASYNC_B32(src, dst)  (*(dst) = *(src))
#define ASYNC_B128(src, dst) (*(v4f*)(dst) = *(const v4f*)(src))
#define ASYNC_WAIT() ((void)0)
#endif

// ---------------------------------------------------------------------------
// Kernel 0: wsum[96*32] = sum over first (r+1) class weight matrices
// ---------------------------------------------------------------------------
__global__ void wsum_kernel(const float* __restrict__ weight,
                            const int* __restrict__ rptr,
                            float* __restrict__ wsum) {
    int j = blockIdx.x * 256 + threadIdx.x;
    if (j >= FIN * HID) return;
    int nc = *rptr + 1;
    if (nc > NCLS) nc = NCLS;
    if (nc < 1)    nc = 1;
    float s = 0.0f;
    for (int c = 0; c < nc; ++c) s += weight[c * FIN * HID + j];
    wsum[j] = s;
}

// ---------------------------------------------------------------------------
// Kernel 1: support[8000][32] = input[8000][96] @ wsum[96][32] + bias
// ---------------------------------------------------------------------------
__global__ __launch_bounds__(256) void support_kernel(
        const float* __restrict__ input, const float* __restrict__ wsum,
        const float* __restrict__ bias, float* __restrict__ support) {
    __shared__ float lin[8 * FIN];
    const int rb = blockIdx.x * 8;
    for (int i = threadIdx.x; i < 8 * FIN; i += 256)
        lin[i] = input[(size_t)rb * FIN + i];
    __syncthreads();
    const int lr = threadIdx.x >> 5;
    const int n  = threadIdx.x & 31;
    float acc = bias[n];
    #pragma unroll 8
    for (int k = 0; k < FIN; ++k)
        acc = fmaf(lin[lr * FIN + k], wsum[k * HID + n], acc);
    support[(size_t)(rb + lr) * HID + n] = acc;
}

// ---------------------------------------------------------------------------
// Stage one K-block of tiles into the given LDS buffer (async when available).
//   which==0 : A = adj[m][k]   (row-contiguous -> b128 async copies)
//   which==1 : A = adj^T       (per-lane transposed async scatter)
// Support tile is always stored transposed [n][k] for the b64 WMMA-B feed.
// ---------------------------------------------------------------------------
__device__ __forceinline__ void stage_tiles(
        int which, int tid, int lane, int half, int mr, int m0, int k0,
        const float* __restrict__ adj, const float* __restrict__ supp,
        float* __restrict__ lB, float* __restrict__ lA) {
    #pragma unroll
    for (int i = 0; i < (BK * HID) / 128; ++i) {
        int idx = tid + i * 128;
        int k = idx >> 5, n = idx & 31;
        ASYNC_B32(&supp[(size_t)(k0 + k) * HID + n], &lB[n * KSTR + k]);
    }
    if (which == 0) {
        const int row = lane >> 1;
        const int hk  = lane & 1;
        int grow = m0 + row; if (grow > U_DIM - 1) grow = U_DIM - 1;
        const float* src = adj + (size_t)grow * U_DIM + k0 + hk * (BK / 2);
        float* dst = &lA[row * KSTR + hk * (BK / 2)];
        #pragma unroll
        for (int i = 0; i < BK / 8; ++i)
            ASYNC_B128(src + 4 * i, dst + 4 * i);
    } else {
        int gcol = m0 + mr; if (gcol > U_DIM - 1) gcol = U_DIM - 1;
        #pragma unroll 8
        for (int j = 0; j < BK / 2; ++j) {
            int r = j * 2 + half;
            ASYNC_B32(&adj[(size_t)(k0 + r) * U_DIM + gcol], &lA[mr * KSTR + r]);
        }
    }
}

// ---------------------------------------------------------------------------
// Kernel 2: the two big GEMMs via V_WMMA_F32_16X16X4_F32, double-buffered LDS
// with ASYNCcnt-tracked global->LDS copies prefetching the next K block.
//   which==0 : out[0:4000]    = deg ⊙ (adj  @ support[4000:8000])
//   which==1 : out[4000:8000] = deg ⊙ (adjT @ support[0:4000])
// ---------------------------------------------------------------------------
__global__ __launch_bounds__(128) void gemm_kernel(
        const float* __restrict__ adj, const float* __restrict__ degree,
        const float* __restrict__ support, float* __restrict__ out) {
    __shared__ __align__(16) float ldsB[2][HID * KSTR];
    __shared__ __align__(16) float ldsA[2][WAVES][16 * KSTR];

    const int which = blockIdx.y;
    const int tid   = threadIdx.x;
    const int lane  = tid & 31;
    const int wave  = tid >> 5;
    const int half  = lane >> 4;     // K=+2 lane group (lanes 16..31)
    const int mr    = lane & 15;     // M for A / N for B
    const int m0    = blockIdx.x * ROWS_PER_WG + wave * 16;

    const float* supp = support + (which == 0 ? (size_t)U_DIM * HID : 0);

    v8f acc0 = {0.f, 0.f, 0.f, 0.f, 0.f, 0.f, 0.f, 0.f};
    v8f acc1 = {0.f, 0.f, 0.f, 0.f, 0.f, 0.f, 0.f, 0.f};

    // prologue: stage block 0 into buffer 0
    stage_tiles(which, tid, lane, half, mr, m0, 0, adj, supp,
                &ldsB[0][0], &ldsA[0][wave][0]);
    ASYNC_WAIT();
    __syncthreads();

    int buf = 0;
    for (int k0 = 0; k0 < U_DIM; k0 += BK) {
        if (k0 + BK < U_DIM)   // async-prefetch next K block into the other buffer
            stage_tiles(which, tid, lane, half, mr, m0, k0 + BK, adj, supp,
                        &ldsB[buf ^ 1][0], &ldsA[buf ^ 1][wave][0]);

        const float* lB = &ldsB[buf][0];
        const float* lA = &ldsA[buf][wave][0];
        #pragma unroll
        for (int kk = 0; kk < BK; kk += 4) {
            v2f a  = *(const v2f*)&lA[mr * KSTR + kk + 2 * half];
            v2f b0 = *(const v2f*)&lB[mr * KSTR + kk + 2 * half];
            v2f b1 = *(const v2f*)&lB[(mr + 16) * KSTR + kk + 2 * half];
            acc0 = __builtin_amdgcn_wmma_f32_16x16x4_f32(
                       false, a, false, b0, (short)0, acc0, false, false);
            acc1 = __builtin_amdgcn_wmma_f32_16x16x4_f32(
                       false, a, false, b1, (short)0, acc1, false, false);
        }
        ASYNC_WAIT();      // next buffer fully written (per-wave ASYNCcnt)
        __syncthreads();   // all waves done reading current buffer
        buf ^= 1;
    }

    // fuse degree row-scale into the store (C/D layout: VGPR i -> M=i, +8 hi half)
    const int degBase = (which == 0) ? 0 : U_DIM;
    #pragma unroll
    for (int i = 0; i < 8; ++i) {
        int row = m0 + i + 8 * half;
        if (row < U_DIM) {
            float d  = degree[degBase + row];
            float* o = out + (size_t)(degBase + row) * HID;
            o[mr]      = d * acc0[i];
            o[mr + 16] = d * acc1[i];
        }
    }
}

// ---------------------------------------------------------------------------
extern "C" void kernel_launch(void* const* d_in, const int* in_sizes, int n_in,
                              void* d_out, int out_size, void* d_ws, size_t ws_size,
                              hipStream_t stream) {
    const float* input  = (const float*)d_in[0];   // [8000, 96]
    const float* adj    = (const float*)d_in[1];   // [4000, 4000]
    const float* degree = (const float*)d_in[2];   // [8000]
    const float* weight = (const float*)d_in[3];   // [5, 96, 32]
    const float* bias   = (const float*)d_in[4];   // [32]
    const int*   rptr   = (const int*)d_in[5];     // scalar r
    float* out = (float*)d_out;                    // [8000, 32]

    float* wsum    = (float*)d_ws;                 // 3072 floats
    float* support = wsum + 4096;                  // 8000*32 floats

    wsum_kernel<<<dim3((FIN * HID + 255) / 256), dim3(256), 0, stream>>>(
        weight, rptr, wsum);
    support_kernel<<<dim3((U_DIM * 2) / 8), dim3(256), 0, stream>>>(
        input, wsum, bias, support);
    dim3 grid((U_DIM + ROWS_PER_WG - 1) / ROWS_PER_WG, 2);
    gemm_kernel<<<grid, dim3(128), 0, stream>>>(adj, degree, support, out);
}